// weight_noise_fn_78271484002473
// MI455X (gfx1250) — compile-verified
//
#include <hip/hip_runtime.h>
#include <hip/hip_bf16.h>
#include <cstdint>

// out[i] = noise[i] * 0.05 * sqrt( spread_bits( trunc(x[i] * 2^14) ) ) / 2^14
// where spread_bits maps bit k -> bit 2k (so each set bit contributes 4^k).
//
// Memory-bound streaming kernel for MI455X (gfx1250, wave32):
//  - float4 (b128) global traffic
//  - x stream double-buffered through LDS via GLOBAL_LOAD_ASYNC_TO_LDS_B128
//    (ASYNCcnt pipeline, s_wait_asynccnt), noise via direct b128 loads.

#define BLOCK 256
#define ITERS_PER_THREAD 8

__device__ __forceinline__ float sigma_of(float xv) {
    // truncation toward zero matches .astype(int32); x in [0,1) -> w < 2^14
    unsigned w = (unsigned)(int)(xv * 16384.0f);
    w &= 0xFFFFu;
    // Part1By1: spread bit k to bit 2k  => value == sum over set bits of 4^k
    w = (w | (w << 8)) & 0x00FF00FFu;
    w = (w | (w << 4)) & 0x0F0F0F0Fu;
    w = (w | (w << 2)) & 0x33333333u;
    w = (w | (w << 1)) & 0x55555555u;
    // 0.05f / 2^14 : division by power of two is exact, so this matches
    // (0.05 * sqrt(acc)) / 16384 bit-for-bit.
    return (0.05f / 16384.0f) * sqrtf((float)w);
}

__global__ __launch_bounds__(BLOCK) void sigma_noise_async_kernel(
    const float4* __restrict__ x4, const float4* __restrict__ nz4,
    float4* __restrict__ o4, unsigned n4)
{
    __shared__ __align__(16) float xtile[2][BLOCK * 4];

    const unsigned tid    = threadIdx.x;
    const unsigned stride = gridDim.x * (unsigned)BLOCK;
    unsigned cur          = blockIdx.x * (unsigned)BLOCK + tid;

    // Wave-relative LDS byte offsets (generic shared addr: offset lives in low 32 bits)
    const unsigned lds0 = (unsigned)(uintptr_t)&xtile[0][tid * 4];
    const unsigned lds1 = (unsigned)(uintptr_t)&xtile[1][tid * 4];

    // Prologue: async-load first x chunk into slot 0 (ASYNCcnt++)
    if (cur < n4) {
        asm volatile("global_load_async_to_lds_b128 %0, %1, off"
                     :: "v"(lds0), "v"(x4 + cur) : "memory");
    }

    unsigned slot = 0u;
    while (cur < n4) {
        const unsigned nxt     = cur + stride;
        const unsigned lds_nxt = slot ? lds0 : lds1;
        const bool     pre     = (nxt < n4);

        // Prefetch next x chunk into the other LDS slot (double buffer)
        if (pre) {
            asm volatile("global_load_async_to_lds_b128 %0, %1, off"
                         :: "v"(lds_nxt), "v"(x4 + nxt) : "memory");
        }

        // Direct b128 load of noise overlaps with the async copies
        float4 nv = nz4[cur];

        // Async ops complete in order per wave: waiting for <=1 outstanding
        // guarantees the chunk for `cur` has landed in LDS.
        if (pre) {
            asm volatile("s_wait_asynccnt 0x1" ::: "memory");
        } else {
            asm volatile("s_wait_asynccnt 0x0" ::: "memory");
        }

        const float4 xv = *(const float4*)&xtile[slot][tid * 4]; // ds_load_b128

        float4 r;
        r.x = nv.x * sigma_of(xv.x);
        r.y = nv.y * sigma_of(xv.y);
        r.z = nv.z * sigma_of(xv.z);
        r.w = nv.w * sigma_of(xv.w);
        o4[cur] = r;                                             // global_store_b128

        cur  = nxt;
        slot ^= 1u;
    }
}

// Handles n % 4 leftover elements (zero for the 4096x8192 shape, kept for generality)
__global__ void sigma_noise_tail_kernel(const float* __restrict__ x,
                                        const float* __restrict__ nz,
                                        float* __restrict__ out,
                                        int start, int n)
{
    int i = start + (int)(blockIdx.x * blockDim.x + threadIdx.x);
    if (i < n) out[i] = nz[i] * sigma_of(x[i]);
}

extern "C" void kernel_launch(void* const* d_in, const int* in_sizes, int n_in,
                              void* d_out, int out_size, void* d_ws, size_t ws_size,
                              hipStream_t stream) {
    (void)n_in; (void)d_ws; (void)ws_size;
    const float* x  = (const float*)d_in[0];
    const float* nz = (const float*)d_in[1];
    float*       out = (float*)d_out;

    const int n  = out_size;      // 4096*8192 = 33,554,432
    const int n4 = n >> 2;        // 8,388,608 float4s

    if (n4 > 0) {
        int blocks = (n4 + BLOCK * ITERS_PER_THREAD - 1) / (BLOCK * ITERS_PER_THREAD);
        if (blocks < 1) blocks = 1;
        sigma_noise_async_kernel<<<blocks, BLOCK, 0, stream>>>(
            (const float4*)x, (const float4*)nz, (float4*)out, (unsigned)n4);
    }
    if (n & 3) {
        sigma_noise_tail_kernel<<<1, 64, 0, stream>>>(x, nz, out, n4 << 2, n);
    }
}